// RobustPPRGo_50070728737388
// MI455X (gfx1250) — compile-verified
//
#include <hip/hip_runtime.h>
#include <hip/hip_bf16.h>

typedef float v2f __attribute__((ext_vector_type(2)));
typedef float v8f __attribute__((ext_vector_type(8)));

#define NFEAT 512
#define NHID  512
#define NCLS  64

#define PANB_STRIDE 516   // W1^T panel row stride (words): dbank=4/lane, conflict-free b64
#define PANB2_STRIDE 68   // W2^T panel row stride
#define HST_STRIDE  68    // h-tile row stride (even => aligned float2, dbank=4)

// ---------- helpers: order-preserving float <-> uint key ----------
__device__ __forceinline__ unsigned f2key(float f) {
    unsigned b = __float_as_uint(f);
    return (b & 0x80000000u) ? ~b : (b | 0x80000000u);
}
__device__ __forceinline__ float key2f(unsigned k) {
    unsigned b = (k & 0x80000000u) ? (k ^ 0x80000000u) : ~k;
    return __uint_as_float(b);
}

// ---------- kernel 0: 32x32 LDS-tiled transpose (dst[c][r] = src[r][c]) ----------
__global__ void __launch_bounds__(256)
transpose_kernel(const float* __restrict__ src, float* __restrict__ dst,
                 int rows, int cols) {
    __shared__ float t[32][33];
    const int bx = blockIdx.x * 32;           // col base
    const int by = blockIdx.y * 32;           // row base
    const int tx = threadIdx.x & 31;
    const int ty = threadIdx.x >> 5;          // 0..7
    #pragma unroll
    for (int i = ty; i < 32; i += 8)
        t[i][tx] = src[(size_t)(by + i) * cols + bx + tx];
    __syncthreads();
    #pragma unroll
    for (int i = ty; i < 32; i += 8)
        dst[(size_t)(bx + i) * rows + by + tx] = t[tx][i];
}

// ---------- kernel 1: sum(ppr) + init radix-select state ----------
__global__ void ppr_sum_init_kernel(const float* __restrict__ ppr, float* wtot,
                                    unsigned* prefix, float* tgt, int n) {
    __shared__ float red[256];
    int tid = threadIdx.x;
    float s = 0.f;
    for (int i = tid; i < n; i += 256) s += ppr[i];
    red[tid] = s;
    __syncthreads();
    for (int off = 128; off > 0; off >>= 1) {
        if (tid < off) red[tid] += red[tid + off];
        __syncthreads();
    }
    float total = red[0];
    if (tid == 0) wtot[0] = total;
    if (tid < NCLS) { prefix[tid] = 0u; tgt[tid] = 0.5f * total; }
}

// ---------- kernel 2: logits = relu(X@W1)@W2 via f32 WMMA ----------
// block = 256 threads = 8 waves; wave owns a 16-row strip (128 rows/block).
// W1^T / W2^T panels staged per pass in LDS and shared by all 8 waves.
__global__ void __launch_bounds__(256)
gemm_logits_kernel(const float* __restrict__ X, const float* __restrict__ W1t,
                   const float* __restrict__ W2t, float* __restrict__ logits) {
    __shared__ float panelB[64 * PANB_STRIDE];     // 132.1 KB : W1^T[ncol0+nloc][k]
    __shared__ float panelB2[64 * PANB2_STRIDE];   //  17.4 KB : W2^T[cls][ncol0+kk]
    __shared__ float hst[8][16 * HST_STRIDE];      //  34.8 KB : per-wave relu(h) tile

    const int tid  = threadIdx.x;
    const int wave = tid >> 5;
    const int lane = tid & 31;
    const int l15  = lane & 15;
    const int kh   = lane >> 4;      // 0: K pair {0,1}, 1: K pair {2,3}

    const size_t row0 = ((size_t)blockIdx.x * 8 + wave) * 16;
    const float* xrow = X + (row0 + (size_t)l15) * NFEAT;

    const v8f vzero = {0.f, 0.f, 0.f, 0.f, 0.f, 0.f, 0.f, 0.f};
    v8f lacc[4];
    #pragma unroll
    for (int j = 0; j < 4; ++j) lacc[j] = vzero;

    for (int p = 0; p < 8; ++p) {                  // 8 N-panels of 64 hidden cols
        const int ncol0 = p * 64;
        __syncthreads();                           // previous panel fully consumed

        // cooperative stage of W1^T panel: 64 rows x 512 k (b128 loads)
        for (int i = tid; i < 64 * 128; i += 256) {
            const int nloc = i >> 7;
            const int k4   = (i & 127) * 4;
            *(float4*)&panelB[nloc * PANB_STRIDE + k4] =
                *(const float4*)&W1t[(size_t)(ncol0 + nloc) * NFEAT + k4];
        }
        // cooperative stage of W2^T panel: 64 classes x 64 k
        for (int i = tid; i < 64 * 16; i += 256) {
            const int nloc = i >> 4;
            const int k4   = (i & 15) * 4;
            *(float4*)&panelB2[nloc * PANB2_STRIDE + k4] =
                *(const float4*)&W2t[(size_t)nloc * NHID + ncol0 + k4];
        }
        __syncthreads();

        v8f hacc[4];
        #pragma unroll
        for (int j = 0; j < 4; ++j) hacc[j] = vzero;

        // GEMM1 panel: h[16x64] += X[16x512] * W1[512, ncol0:ncol0+64]
        #pragma unroll 2
        for (int k0 = 0; k0 < NFEAT; k0 += 4) {
            const int kb = k0 + 2 * kh;
            const v2f a = *(const v2f*)(xrow + kb);            // global b64
            #pragma unroll
            for (int j = 0; j < 4; ++j) {
                const v2f b = *(const v2f*)&panelB[(j * 16 + l15) * PANB_STRIDE + kb];
                hacc[j] = __builtin_amdgcn_wmma_f32_16x16x4_f32(
                    false, a, false, b, (short)0, hacc[j], false, false);
            }
        }

        // ReLU + C-layout -> per-wave LDS tile (row-major => GEMM2 A-layout source)
        #pragma unroll
        for (int j = 0; j < 4; ++j) {
            #pragma unroll
            for (int v = 0; v < 8; ++v) {
                float val = hacc[j][v];
                hst[wave][(v + 8 * kh) * HST_STRIDE + j * 16 + l15] =
                    val > 0.f ? val : 0.f;
            }
        }
        // same-wave LDS ops are in-order: no barrier needed before reading hst

        // GEMM2 partial: logits[16x64] += h[16x64] * W2[ncol0:ncol0+64, 0:64]
        #pragma unroll 2
        for (int kk = 0; kk < 64; kk += 4) {
            const int kb = kk + 2 * kh;
            const v2f a = *(const v2f*)&hst[wave][l15 * HST_STRIDE + kb];
            #pragma unroll
            for (int j = 0; j < 4; ++j) {
                const v2f b = *(const v2f*)&panelB2[(j * 16 + l15) * PANB2_STRIDE + kb];
                lacc[j] = __builtin_amdgcn_wmma_f32_16x16x4_f32(
                    false, a, false, b, (short)0, lacc[j], false, false);
            }
        }
    }

    // store logits tile (C layout: vgpr v -> row v + 8*kh, lane -> col)
    #pragma unroll
    for (int j = 0; j < 4; ++j) {
        #pragma unroll
        for (int v = 0; v < 8; ++v) {
            logits[(row0 + v + 8 * kh) * NCLS + j * 16 + l15] = lacc[j][v];
        }
    }
}

// ---------- kernel 3: weighted per-column histogram for radix select ----------
__global__ void hist_kernel(const float* __restrict__ logits, const float* __restrict__ ppr,
                            const unsigned* __restrict__ prefix, float* __restrict__ hist,
                            int shift, int n) {
    __shared__ float    lh[NCLS * 256];
    __shared__ unsigned pref[NCLS];
    const int tid = threadIdx.x;
    for (int i = tid; i < NCLS * 256; i += blockDim.x) lh[i] = 0.f;
    if (tid < NCLS) pref[tid] = prefix[tid];
    __syncthreads();

    const unsigned maskHigh = (shift >= 24) ? 0u : (0xFFFFFFFFu << (shift + 8));
    const size_t total = (size_t)n * NCLS;
    for (size_t idx = (size_t)blockIdx.x * blockDim.x + tid; idx < total;
         idx += (size_t)gridDim.x * blockDim.x) {
        const int c = (int)(idx & (NCLS - 1));
        const unsigned key = f2key(logits[idx]);
        if (((key ^ pref[c]) & maskHigh) == 0u) {
            const unsigned b = (key >> shift) & 0xFFu;
            atomicAdd(&lh[c * 256 + (int)b], ppr[idx >> 6]);
        }
    }
    __syncthreads();
    for (int i = tid; i < NCLS * 256; i += blockDim.x)
        if (lh[i] != 0.f) atomicAdd(&hist[i], lh[i]);
}

// ---------- kernel 4: pick bucket where cumulative weight crosses target ----------
__global__ void select_kernel(const float* __restrict__ hist, unsigned* prefix,
                              float* tgt, float* median, int shift) {
    const int c = threadIdx.x;
    if (c >= NCLS) return;
    const float t = tgt[c];
    float cum = 0.f, cumBefore = 0.f, cumBeforeNZ = 0.f;
    int chosen = 255, lastNZ = 0;
    bool found = false;
    for (int b = 0; b < 256; ++b) {
        const float h = hist[c * 256 + b];
        if (h > 0.f) { lastNZ = b; cumBeforeNZ = cum; }
        if (!found && h > 0.f && cum + h >= t) { chosen = b; cumBefore = cum; found = true; }
        cum += h;
    }
    if (!found) { chosen = lastNZ; cumBefore = cumBeforeNZ; }
    prefix[c] |= ((unsigned)chosen) << shift;
    tgt[c] = t - cumBefore;
    if (shift == 0) median[c] = key2f(prefix[c]);
}

// ---------- kernel 5: distances + softmax-weighted accumulation ----------
__global__ void __launch_bounds__(256)
dist_accum_kernel(const float* __restrict__ logits, const float* __restrict__ ppr,
                  const float* __restrict__ median, float* S0, float* S1) {
    __shared__ float tile[256 * 65];  // padded: no LDS bank conflicts on row reads
    __shared__ float parr[256];
    __shared__ float medS[NCLS];
    const int tid = threadIdx.x;
    const size_t rowBase = (size_t)blockIdx.x * 256;
    const float* src = logits + rowBase * NCLS;

    for (int k = tid; k < 256 * NCLS; k += 256) {
        const int r = k >> 6, c = k & 63;
        tile[r * 65 + c] = src[k];
    }
    if (tid < NCLS) medS[tid] = median[tid];
    __syncthreads();

    {   // one row per thread
        float ssq = 0.f;
        #pragma unroll
        for (int c = 0; c < NCLS; ++c) {
            const float d = tile[tid * 65 + c] - medS[c];
            ssq += d * d;
        }
        const float dist = sqrtf(ssq) * 0.125f;   // / sqrt(64)
        parr[tid] = ppr[rowBase + tid] * expf(-dist);
    }
    __syncthreads();

    if (tid < NCLS) {   // per-class weighted column sum
        float s = 0.f;
        for (int r = 0; r < 256; ++r) s += parr[r] * tile[r * 65 + tid];
        atomicAdd(&S1[tid], s);
    }
    __syncthreads();
    for (int off = 128; off > 0; off >>= 1) {
        if (tid < off) parr[tid] += parr[tid + off];
        __syncthreads();
    }
    if (tid == 0) atomicAdd(S0, parr[0]);
}

// ---------- kernel 6: final 64 outputs ----------
__global__ void finalize_kernel(const float* S0, const float* S1, const float* wtot,
                                float* __restrict__ out) {
    const int c = threadIdx.x;
    if (c < NCLS) out[c] = wtot[0] * S1[c] / S0[0];
}

extern "C" void kernel_launch(void* const* d_in, const int* in_sizes, int n_in,
                              void* d_out, int out_size, void* d_ws, size_t ws_size,
                              hipStream_t stream) {
    const float* X   = (const float*)d_in[0];
    const float* ppr = (const float*)d_in[1];
    const float* W1  = (const float*)d_in[2];
    const float* W2  = (const float*)d_in[3];
    float* out = (float*)d_out;
    const int N = in_sizes[1];  // 131072 neighbors

    // workspace layout (all fp32 / u32, 4B aligned)
    float*    logits = (float*)d_ws;                   // N * 64
    float*    hist   = logits + (size_t)N * NCLS;      // 64 * 256
    unsigned* prefix = (unsigned*)(hist + NCLS * 256); // 64
    float*    tgt    = (float*)(prefix + NCLS);        // 64
    float*    median = tgt + NCLS;                     // 64
    float*    S0     = median + NCLS;                  // 1
    float*    S1     = S0 + 1;                         // 64
    float*    wtot   = S1 + NCLS;                      // 1
    float*    W1t    = wtot + 1;                       // 512*512  (W1^T[n][k])
    float*    W2t    = W1t + (size_t)NFEAT * NHID;     // 64*512   (W2^T[c][k])

    // 0) k-contiguous transposed weights (1.1 MB once per call)
    transpose_kernel<<<dim3(NHID / 32, NFEAT / 32), 256, 0, stream>>>(W1, W1t, NFEAT, NHID);
    transpose_kernel<<<dim3(NCLS / 32, NHID / 32), 256, 0, stream>>>(W2, W2t, NHID, NCLS);

    // 1) total PPR weight + init select state
    ppr_sum_init_kernel<<<1, 256, 0, stream>>>(ppr, wtot, prefix, tgt, N);

    // 2) logits = relu(X@W1)@W2  (f32 WMMA, LDS-staged weight panels,
    //    memory-bound on the 268MB X read)
    gemm_logits_kernel<<<N / 128, 256, 0, stream>>>(X, W1t, W2t, logits);

    // 3) exact weighted median per class via 4-pass radix select on float keys
    for (int pass = 0; pass < 4; ++pass) {
        const int shift = 24 - 8 * pass;
        hipMemsetAsync(hist, 0, NCLS * 256 * sizeof(float), stream);
        hist_kernel<<<128, 256, 0, stream>>>(logits, ppr, prefix, hist, shift, N);
        select_kernel<<<1, 64, 0, stream>>>(hist, prefix, tgt, median, shift);
    }

    // 4) distances -> exp weights -> weighted logit sum
    hipMemsetAsync(S0, 0, (1 + NCLS) * sizeof(float), stream);
    dist_accum_kernel<<<N / 256, 256, 0, stream>>>(logits, ppr, median, S0, S1);

    // 5) out[c] = W * sum(p*logit_c) / sum(p)
    finalize_kernel<<<1, 64, 0, stream>>>(S0, S1, wtot, out);
}